// SE3Transformer_73220602462722
// MI455X (gfx1250) — compile-verified
//
#include <hip/hip_runtime.h>
#include <math.h>

#define NN 16000
#define NE 256000
#define NG 160

typedef float v2f __attribute__((ext_vector_type(2)));
typedef float v8f __attribute__((ext_vector_type(8)));

__device__ __forceinline__ v8f wmma4(v2f a, v2f b, v8f c) {
  // D = A(16x4,f32) * B(4x16,f32) + C(16x16,f32)
  return __builtin_amdgcn_wmma_f32_16x16x4_f32(false, a, false, b, (short)0, c, false, false);
}

// ---------------------------------------------------------------- basis ----
__global__ void basis_kernel(const float* __restrict__ pos,
                             const float* __restrict__ efeat,
                             const int* __restrict__ src,
                             const int* __restrict__ dst,
                             float* __restrict__ xer,
                             float* __restrict__ K01,
                             float* __restrict__ K11) {
  int e = blockIdx.x * blockDim.x + threadIdx.x;
  if (e >= NE) return;
  int s = src[e], d = dst[e];
  float rx = pos[d * 3 + 0] - pos[s * 3 + 0];
  float ry = pos[d * 3 + 1] - pos[s * 3 + 1];
  float rz = pos[d * 3 + 2] - pos[s * 3 + 2];
  float r = sqrtf(rx * rx + ry * ry + rz * rz + 1e-12f);
  float up[3] = {ry / r, rz / r, rx / r};  // u permuted by P=[1,2,0]
  const float Y0 = 0.28209479177387814f;
  const float c1 = 0.4886025119029199f;
  const float c2 = 1.0925484305920792f;
  xer[e * 5 + 0] = efeat[e * 4 + 0];
  xer[e * 5 + 1] = efeat[e * 4 + 1];
  xer[e * 5 + 2] = efeat[e * 4 + 2];
  xer[e * 5 + 3] = efeat[e * 4 + 3];
  xer[e * 5 + 4] = r;
  K01[e * 3 + 0] = c1 * up[0];
  K01[e * 3 + 1] = c1 * up[1];
  K01[e * 3 + 2] = c1 * up[2];
  const float j0 = Y0 * 0.5773502691896258f;  // Y0/sqrt(3)
  const float s1 = c1 * 0.7071067811865475f;  // c1/sqrt(2)
  const float s2 = c2 * 0.7071067811865475f;  // c2/sqrt(2)
  float* K = K11 + (long)e * 27;
#pragma unroll
  for (int a = 0; a < 3; a++) {
#pragma unroll
    for (int b = 0; b < 3; b++) {
      float J0v = (a == b) ? j0 : 0.f;
      float J1v = 0.f;
      if (a != b) {
        int cc = 3 - a - b;
        float sg = (((b - a + 3) % 3) == 1) ? 1.f : -1.f;  // eps_{abc}
        J1v = s1 * sg * up[cc];
      }
      float J2v = s2 * (up[a] * up[b] - ((a == b) ? (1.f / 3.f) : 0.f));
      K[(a * 3 + b) * 3 + 0] = J0v;
      K[(a * 3 + b) * 3 + 1] = J1v;
      K[(a * 3 + b) * 3 + 2] = J2v;
    }
  }
}

// ------------------------------------------- fused radial MLP + conv ------
// One wave per 16-edge tile. MLP: 5->32 (scalar), 32->32 and 32->KOUT via
// V_WMMA_F32_16X16X4_F32 (M=16 edges, N=16-col tiles, K chunks of 4).
template <int CI, int CO, int NJ, int MI, int MO>
__global__ __launch_bounds__(32) void edge_conv_kernel(
    const float* __restrict__ xer, const float* __restrict__ K01g,
    const float* __restrict__ K11g, const float* __restrict__ hsrc,
    const int* __restrict__ src,
    const float* __restrict__ W1, const float* __restrict__ b1,
    const float* __restrict__ W2, const float* __restrict__ b2,
    const float* __restrict__ W3, const float* __restrict__ b3,
    float* __restrict__ out, int ostride, int col0, int accum) {
  constexpr int KOUT = CO * CI * NJ;
  constexpr int NT = KOUT / 16;
  __shared__ float sx[16][5];
  __shared__ float sh1[16][33];
  __shared__ float sh2[16][33];
  __shared__ float sR[16][KOUT + 1];
  const int lane = threadIdx.x;
  const int half = lane >> 4;
  const int idx = lane & 15;
  const long e0 = (long)blockIdx.x * 16;

  for (int t = lane; t < 80; t += 32) sx[t / 5][t % 5] = xer[e0 * 5 + t];
  __syncthreads();

  {  // layer 1: lane computes h1 column `lane` for all 16 edges
    float w0 = W1[lane], w1 = W1[32 + lane], w2 = W1[64 + lane];
    float w3 = W1[96 + lane], w4 = W1[128 + lane];
    float bb = b1[lane];
#pragma unroll
    for (int m = 0; m < 16; m++) {
      float s = bb + sx[m][0] * w0 + sx[m][1] * w1 + sx[m][2] * w2 +
                sx[m][3] * w3 + sx[m][4] * w4;
      sh1[m][lane] = fmaxf(s, 0.f);
    }
  }
  __syncthreads();

#pragma unroll
  for (int nt = 0; nt < 2; nt++) {  // layer 2: 16x32 = relu(h1 @ W2 + b2)
    const int nc = nt * 16 + idx;
    v8f acc;
    const float bb = b2[nc];
#pragma unroll
    for (int r = 0; r < 8; r++) acc[r] = bb;
#pragma unroll
    for (int k0 = 0; k0 < 32; k0 += 4) {
      v2f a, b;
      a[0] = sh1[idx][k0 + 2 * half];
      a[1] = sh1[idx][k0 + 2 * half + 1];
      b[0] = W2[(k0 + 2 * half) * 32 + nc];
      b[1] = W2[(k0 + 2 * half + 1) * 32 + nc];
      acc = wmma4(a, b, acc);
    }
#pragma unroll
    for (int r = 0; r < 8; r++) sh2[r + 8 * half][nc] = fmaxf(acc[r], 0.f);
  }
  __syncthreads();

  for (int nt = 0; nt < NT; nt++) {  // layer 3: 16xKOUT = h2 @ W3 + b3
    const int nc = nt * 16 + idx;
    v8f acc;
    const float bb = b3[nc];
#pragma unroll
    for (int r = 0; r < 8; r++) acc[r] = bb;
#pragma unroll
    for (int k0 = 0; k0 < 32; k0 += 4) {
      v2f a, b;
      a[0] = sh2[idx][k0 + 2 * half];
      a[1] = sh2[idx][k0 + 2 * half + 1];
      b[0] = W3[(k0 + 2 * half) * KOUT + nc];
      b[1] = W3[(k0 + 2 * half + 1) * KOUT + nc];
      acc = wmma4(a, b, acc);
    }
#pragma unroll
    for (int r = 0; r < 8; r++) sR[r + 8 * half][nc] = acc[r];
  }
  __syncthreads();

  {  // equivariant contraction: lane handles edge `idx`, half of CO channels
    const long e = e0 + idx;
    const int s = src[e];
    float fs[CI * MI];
#pragma unroll
    for (int t = 0; t < CI * MI; t++) fs[t] = hsrc[(long)s * (CI * MI) + t];
    float bk[3] = {0.f, 0.f, 0.f};
    if (MI == 3 || MO == 3) {
      bk[0] = K01g[e * 3 + 0];
      bk[1] = K01g[e * 3 + 1];
      bk[2] = K01g[e * 3 + 2];
    }
    float bK[27];
    if (MI == 3 && MO == 3) {
#pragma unroll
      for (int t = 0; t < 27; t++) bK[t] = K11g[e * 27 + t];
    }
    constexpr int COH = CO / 2;
    for (int dd = 0; dd < COH; dd++) {
      const int d = half * COH + dd;
      float m0 = 0.f, m1 = 0.f, m2 = 0.f;
      if (MI == 1 && MO == 1) {
        float t = 0.f;
#pragma unroll
        for (int c = 0; c < CI; c++) t += sR[idx][d * CI + c] * fs[c];
        m0 = 0.28209479177387814f * t;
      } else if (MI == 1 && MO == 3) {
        float t = 0.f;
#pragma unroll
        for (int c = 0; c < CI; c++) t += sR[idx][d * CI + c] * fs[c];
        m0 = bk[0] * t; m1 = bk[1] * t; m2 = bk[2] * t;
      } else if (MI == 3 && MO == 1) {
        float t = 0.f;
#pragma unroll
        for (int c = 0; c < CI; c++) {
          float g = bk[0] * fs[c * 3] + bk[1] * fs[c * 3 + 1] + bk[2] * fs[c * 3 + 2];
          t += sR[idx][d * CI + c] * g;
        }
        m0 = t;
      } else {  // 1,1 pair: NJ=3
        float tmp[9];
#pragma unroll
        for (int t = 0; t < 9; t++) tmp[t] = 0.f;
#pragma unroll
        for (int c = 0; c < CI; c++) {
#pragma unroll
          for (int j = 0; j < 3; j++) {
            float r = sR[idx][(d * CI + c) * 3 + j];
            tmp[0 * 3 + j] += r * fs[c * 3 + 0];
            tmp[1 * 3 + j] += r * fs[c * 3 + 1];
            tmp[2 * 3 + j] += r * fs[c * 3 + 2];
          }
        }
        float ac[3] = {0.f, 0.f, 0.f};
#pragma unroll
        for (int o = 0; o < 3; o++) {
          float sacc = 0.f;
#pragma unroll
          for (int i = 0; i < 3; i++)
#pragma unroll
            for (int j = 0; j < 3; j++) sacc += tmp[i * 3 + j] * bK[(o * 3 + i) * 3 + j];
          ac[o] = sacc;
        }
        m0 = ac[0]; m1 = ac[1]; m2 = ac[2];
      }
      float* op = out + e * ostride + col0 + d * MO;
      if (accum) {
        op[0] += m0;
        if (MO == 3) { op[1] += m1; op[2] += m2; }
      } else {
        op[0] = m0;
        if (MO == 3) { op[1] = m1; op[2] = m2; }
      }
    }
  }
}

// --------------------------------------------------------------- queries ---
__global__ void qn_kernel(const float* __restrict__ q0, const float* __restrict__ q1,
                          const float* __restrict__ h0, const float* __restrict__ h1,
                          float* __restrict__ qn, int CI) {
  int n = blockIdx.x * blockDim.x + threadIdx.x;
  if (n >= NN) return;
  for (int c = 0; c < 4; c++) {
    float s = 0.f;
    for (int k = 0; k < CI; k++) s += q0[c * CI + k] * h0[(long)n * CI + k];
    qn[(long)n * 16 + c] = s;
  }
  if (q1) {
    for (int c = 0; c < 4; c++)
      for (int o = 0; o < 3; o++) {
        float s = 0.f;
        for (int k = 0; k < 16; k++) s += q1[c * 16 + k] * h1[((long)n * 16 + k) * 3 + o];
        qn[(long)n * 16 + 4 + c * 3 + o] = s;
      }
  } else {
    for (int t = 4; t < 16; t++) qn[(long)n * 16 + t] = 0.f;
  }
}

__global__ void fill_kernel(float* __restrict__ p, int n, float v) {
  int i = blockIdx.x * blockDim.x + threadIdx.x;
  if (i < n) p[i] = v;
}

__device__ __forceinline__ void atomicMaxF(float* addr, float v) {
  if (v >= 0.f) atomicMax((int*)addr, __float_as_int(v));
  else atomicMin((unsigned int*)addr, __float_as_uint(v));
}

__global__ void logits_kernel(const float* __restrict__ kmsg, const float* __restrict__ qn,
                              const int* __restrict__ dst, float* __restrict__ att,
                              float* __restrict__ mmax, int D) {
  int t = blockIdx.x * blockDim.x + threadIdx.x;
  if (t >= NE * 2) return;
  int e = t >> 1, h = t & 1;
  const float* kf = kmsg + (long)e * 16;
  const float* qf = qn + (long)dst[e] * 16;
  float s = kf[2 * h] * qf[2 * h] + kf[2 * h + 1] * qf[2 * h + 1];
  if (D == 8) {
#pragma unroll
    for (int u = 0; u < 6; u++) s += kf[4 + 6 * h + u] * qf[4 + 6 * h + u];
  }
  s *= (D == 8) ? 0.35355339059327373f : 0.70710678118654752f;  // 1/sqrt(D)
  att[t] = s;
  atomicMaxF(&mmax[(long)dst[e] * 2 + h], s);
}

__global__ void exp_kernel(const int* __restrict__ dst, float* __restrict__ att,
                           const float* __restrict__ mmax, float* __restrict__ den) {
  int t = blockIdx.x * blockDim.x + threadIdx.x;
  if (t >= NE * 2) return;
  int e = t >> 1, h = t & 1;
  float ex = expf(att[t] - mmax[(long)dst[e] * 2 + h]);
  att[t] = ex;
  atomicAdd(&den[(long)dst[e] * 2 + h], ex);
}

__global__ void zacc_kernel(const float* __restrict__ att, const float* __restrict__ den,
                            const float* __restrict__ vmsg, const int* __restrict__ dst,
                            float* __restrict__ z) {
  int t = blockIdx.x * blockDim.x + threadIdx.x;
  if (t >= NE * 16) return;
  int e = t >> 4, c = t & 15;
  int h = (c < 4) ? (c >> 1) : (((c - 4) / 3) >> 1);
  int d = dst[e];
  float a = att[(long)e * 2 + h] / (den[(long)d * 2 + h] + 1e-8f);
  atomicAdd(&z[(long)d * 16 + c], a * vmsg[(long)e * 16 + c]);
}

// -------------------------------------- proj + residual + gated norm ------
template <int CI, bool HAS1>
__global__ void node_update_kernel(
    const float* __restrict__ z, const float* __restrict__ h0in,
    const float* __restrict__ h1in, const float* __restrict__ proj0,
    const float* __restrict__ proj1, const float* __restrict__ nw0,
    const float* __restrict__ nb0, const float* __restrict__ nw1,
    const float* __restrict__ nb1, float* __restrict__ h0out,
    float* __restrict__ h1out) {
  int n = blockIdx.x * blockDim.x + threadIdx.x;
  if (n >= NN) return;
  const float* zp = z + (long)n * 16;
  float y0[16];
  for (int c = 0; c < 16; c++) {
    float s = 0.f;
    for (int k = 0; k < 4; k++) s += proj0[c * (4 + CI) + k] * zp[k];
    for (int k = 0; k < CI; k++) s += proj0[c * (4 + CI) + 4 + k] * h0in[(long)n * CI + k];
    y0[c] = s + h0in[(long)n * CI + c];  // residual on first 16 channels
  }
  float y1[48];
  if (HAS1) {
    for (int c = 0; c < 16; c++)
      for (int o = 0; o < 3; o++) {
        float s = 0.f;
        for (int k = 0; k < 4; k++) s += proj1[c * 20 + k] * zp[4 + k * 3 + o];
        for (int k = 0; k < 16; k++) s += proj1[c * 20 + 4 + k] * h1in[((long)n * 16 + k) * 3 + o];
        y1[c * 3 + o] = s + h1in[((long)n * 16 + c) * 3 + o];
      }
  } else {
    for (int c = 0; c < 16; c++)
      for (int o = 0; o < 3; o++) {
        float s = 0.f;
        for (int k = 0; k < 4; k++) s += proj1[c * 4 + k] * zp[4 + k * 3 + o];
        y1[c * 3 + o] = s;
      }
  }
  float nn0[16];
  for (int c = 0; c < 16; c++) nn0[c] = sqrtf(y0[c] * y0[c] + 1e-12f);
  for (int c = 0; c < 16; c++) {
    float s = nb0[c];
    for (int k = 0; k < 16; k++) s += nn0[k] * nw0[k * 16 + c];
    s = fmaxf(s, 0.f);
    h0out[(long)n * 16 + c] = s * y0[c] / nn0[c];
  }
  float nn1[16];
  for (int c = 0; c < 16; c++)
    nn1[c] = sqrtf(y1[c * 3] * y1[c * 3] + y1[c * 3 + 1] * y1[c * 3 + 1] +
                   y1[c * 3 + 2] * y1[c * 3 + 2] + 1e-12f);
  for (int c = 0; c < 16; c++) {
    float s = nb1[c];
    for (int k = 0; k < 16; k++) s += nn1[k] * nw1[k * 16 + c];
    s = fmaxf(s, 0.f);
    float inv = s / nn1[c];
    h1out[((long)n * 16 + c) * 3 + 0] = inv * y1[c * 3 + 0];
    h1out[((long)n * 16 + c) * 3 + 1] = inv * y1[c * 3 + 1];
    h1out[((long)n * 16 + c) * 3 + 2] = inv * y1[c * 3 + 2];
  }
}

// ----------------------------------------------------------- final stage ---
__global__ void edge_agg_kernel(const float* __restrict__ fmsg, const int* __restrict__ dst,
                                float* __restrict__ agg) {
  int t = blockIdx.x * blockDim.x + threadIdx.x;
  if (t >= NE * 32) return;
  int e = t >> 5, c = t & 31;
  atomicAdd(&agg[(long)dst[e] * 32 + c], fmsg[(long)e * 32 + c]);
}

__global__ void self_kernel(const float* __restrict__ selfW, const float* __restrict__ h0,
                            float* __restrict__ agg) {
  int t = blockIdx.x * blockDim.x + threadIdx.x;
  if (t >= NN * 32) return;
  int n = t >> 5, c = t & 31;
  float s = 0.f;
  for (int k = 0; k < 16; k++) s += selfW[c * 16 + k] * h0[(long)n * 16 + k];
  agg[(long)n * 32 + c] += s;
}

__global__ void pool_kernel(const float* __restrict__ agg, const int* __restrict__ n2g,
                            float* __restrict__ gsum, float* __restrict__ gcnt) {
  int t = blockIdx.x * blockDim.x + threadIdx.x;
  if (t >= NN * 32) return;
  int n = t >> 5, c = t & 31;
  int g = n2g[n];
  atomicAdd(&gsum[(long)g * 32 + c], agg[(long)n * 32 + c]);
  if (c == 0) atomicAdd(&gcnt[g], 1.f);
}

__global__ void head_kernel(const float* __restrict__ gsum, const float* __restrict__ gcnt,
                            const float* __restrict__ W1, const float* __restrict__ b1v,
                            const float* __restrict__ W2, const float* __restrict__ b2v,
                            float* __restrict__ out) {
  int g = blockIdx.x * blockDim.x + threadIdx.x;
  if (g >= NG) return;
  float inv = 1.f / (gcnt[g] + 1e-8f);
  float x[32];
  for (int c = 0; c < 32; c++) x[c] = gsum[(long)g * 32 + c] * inv;
  float h[32];
  for (int c = 0; c < 32; c++) {
    float s = b1v[c];
    for (int k = 0; k < 32; k++) s += x[k] * W1[k * 32 + c];
    h[c] = fmaxf(s, 0.f);
  }
  float o = b2v[0];
  for (int k = 0; k < 32; k++) o += h[k] * W2[k];
  out[g] = o;
}

// ----------------------------------------------------------------- host ----
extern "C" void kernel_launch(void* const* d_in, const int* in_sizes, int n_in,
                              void* d_out, int out_size, void* d_ws, size_t ws_size,
                              hipStream_t stream) {
  (void)in_sizes; (void)out_size; (void)ws_size;
  const float* pos = (const float*)d_in[0];
  const float* feat = (const float*)d_in[1];
  const float* efeat = (const float*)d_in[2];

  // params pytree leaves in jax order (dicts sorted by key, lists in order)
  const int NP = 210;
  int sz[NP]; int t = 0;
  auto rad = [&](int K) { sz[t++] = 160; sz[t++] = 1024; sz[t++] = 32 * K;
                          sz[t++] = 32; sz[t++] = 32; sz[t++] = K; };
  sz[t++] = 1024; sz[t++] = 32;         // fc1 (W,b)
  sz[t++] = 32; sz[t++] = 1;            // fc2 (W,b)
  rad(512); rad(512); sz[t++] = 512;    // final: pairs '0,0','1,0', self '0'
  rad(80); sz[t++] = 16 * 24; sz[t++] = 16 * 4; sz[t++] = 4 * 20;  // L0 k,proj,q
  rad(80); rad(80);                                                 // L0 v
  for (int l = 1; l < 4; l++) {
    rad(64); rad(64); rad(64); rad(192);          // k
    sz[t++] = 320; sz[t++] = 320;                 // proj
    sz[t++] = 64; sz[t++] = 64;                   // q
    rad(64); rad(64); rad(64); rad(192);          // v
  }
  for (int l = 0; l < 4; l++) { sz[t++] = 256; sz[t++] = 16; sz[t++] = 256; sz[t++] = 16; }

  const float* prm[NP];
  const int *src, *dst, *n2g;
  if (n_in >= 3 + NP + 3) {  // one leaf per slot
    for (int j = 0; j < NP; j++) prm[j] = (const float*)d_in[3 + j];
    src = (const int*)d_in[3 + NP];
    dst = (const int*)d_in[4 + NP];
    n2g = (const int*)d_in[5 + NP];
  } else {  // single concatenated params buffer
    const float* base = (const float*)d_in[3];
    long off = 0;
    for (int j = 0; j < NP; j++) { prm[j] = base + off; off += sz[j]; }
    src = (const int*)d_in[n_in - 3];
    dst = (const int*)d_in[n_in - 2];
    n2g = (const int*)d_in[n_in - 1];
  }

  // workspace carve-up (floats)
  float* w = (float*)d_ws; long o = 0;
  float* xer = w + o;  o += (long)NE * 5;
  float* K01 = w + o;  o += (long)NE * 3;
  float* K11 = w + o;  o += (long)NE * 27;
  float* kmsg = w + o; o += (long)NE * 16;
  float* vmsg = w + o; o += (long)NE * 16;  // fmsg (E x 32) reuses kmsg+vmsg
  float* att = w + o;  o += (long)NE * 2;
  float* qn = w + o;   o += (long)NN * 16;
  float* mmax = w + o; o += (long)NN * 2;
  float* den = w + o;  o += (long)NN * 2;
  float* zbuf = w + o; o += (long)NN * 16;
  float* h0a = w + o;  o += (long)NN * 16;
  float* h1a = w + o;  o += (long)NN * 48;
  float* h0b = w + o;  o += (long)NN * 16;
  float* h1b = w + o;  o += (long)NN * 48;
  float* agg = w + o;  o += (long)NN * 32;
  float* gsum = w + o; o += (long)NG * 32;
  float* gcnt = w + o; o += (long)NG;
  float* fmsg = kmsg;
  (void)vmsg;

  const int T = 256;
  const int EB = NE / 16;
  basis_kernel<<<(NE + T - 1) / T, T, 0, stream>>>(pos, efeat, src, dst, xer, K01, K11);

#define RAD_ARGS(j) prm[j], prm[(j) + 3], prm[(j) + 1], prm[(j) + 4], prm[(j) + 2], prm[(j) + 5]

  const float* h0 = feat;
  const float* h1 = nullptr;
  for (int l = 0; l < 4; l++) {
    const int B = (l == 0) ? 0 : 38 + (l - 1) * 52;
    if (l == 0) {
      edge_conv_kernel<20, 4, 1, 1, 1><<<EB, 32, 0, stream>>>(xer, K01, K11, h0, src, RAD_ARGS(17), kmsg, 16, 0, 0);
      edge_conv_kernel<20, 4, 1, 1, 1><<<EB, 32, 0, stream>>>(xer, K01, K11, h0, src, RAD_ARGS(26), vmsg, 16, 0, 0);
      edge_conv_kernel<20, 4, 1, 1, 3><<<EB, 32, 0, stream>>>(xer, K01, K11, h0, src, RAD_ARGS(32), vmsg, 16, 4, 0);
      qn_kernel<<<(NN + T - 1) / T, T, 0, stream>>>(prm[25], nullptr, h0, nullptr, qn, 20);
    } else {
      edge_conv_kernel<16, 4, 1, 1, 1><<<EB, 32, 0, stream>>>(xer, K01, K11, h0, src, RAD_ARGS(B + 0), kmsg, 16, 0, 0);
      edge_conv_kernel<16, 4, 1, 1, 3><<<EB, 32, 0, stream>>>(xer, K01, K11, h0, src, RAD_ARGS(B + 6), kmsg, 16, 4, 0);
      edge_conv_kernel<16, 4, 1, 3, 1><<<EB, 32, 0, stream>>>(xer, K01, K11, h1, src, RAD_ARGS(B + 12), kmsg, 16, 0, 1);
      edge_conv_kernel<16, 4, 3, 3, 3><<<EB, 32, 0, stream>>>(xer, K01, K11, h1, src, RAD_ARGS(B + 18), kmsg, 16, 4, 1);
      edge_conv_kernel<16, 4, 1, 1, 1><<<EB, 32, 0, stream>>>(xer, K01, K11, h0, src, RAD_ARGS(B + 28), vmsg, 16, 0, 0);
      edge_conv_kernel<16, 4, 1, 1, 3><<<EB, 32, 0, stream>>>(xer, K01, K11, h0, src, RAD_ARGS(B + 34), vmsg, 16, 4, 0);
      edge_conv_kernel<16, 4, 1, 3, 1><<<EB, 32, 0, stream>>>(xer, K01, K11, h1, src, RAD_ARGS(B + 40), vmsg, 16, 0, 1);
      edge_conv_kernel<16, 4, 3, 3, 3><<<EB, 32, 0, stream>>>(xer, K01, K11, h1, src, RAD_ARGS(B + 46), vmsg, 16, 4, 1);
      qn_kernel<<<(NN + T - 1) / T, T, 0, stream>>>(prm[B + 26], prm[B + 27], h0, h1, qn, 16);
    }
    fill_kernel<<<(NN * 2 + T - 1) / T, T, 0, stream>>>(mmax, NN * 2, -INFINITY);
    hipMemsetAsync(den, 0, (size_t)NN * 2 * sizeof(float), stream);
    hipMemsetAsync(zbuf, 0, (size_t)NN * 16 * sizeof(float), stream);
    const int D = (l == 0) ? 2 : 8;
    logits_kernel<<<(NE * 2 + T - 1) / T, T, 0, stream>>>(kmsg, qn, dst, att, mmax, D);
    exp_kernel<<<(NE * 2 + T - 1) / T, T, 0, stream>>>(dst, att, mmax, den);
    zacc_kernel<<<(NE * 16 + T - 1) / T, T, 0, stream>>>(att, den, vmsg, dst, zbuf);
    float* h0o = (l & 1) ? h0b : h0a;
    float* h1o = (l & 1) ? h1b : h1a;
    const int nb = 194 + 4 * l;
    if (l == 0)
      node_update_kernel<20, false><<<(NN + T - 1) / T, T, 0, stream>>>(
          zbuf, h0, nullptr, prm[23], prm[24], prm[nb], prm[nb + 1], prm[nb + 2], prm[nb + 3], h0o, h1o);
    else
      node_update_kernel<16, true><<<(NN + T - 1) / T, T, 0, stream>>>(
          zbuf, h0, h1, prm[B + 24], prm[B + 25], prm[nb], prm[nb + 1], prm[nb + 2], prm[nb + 3], h0o, h1o);
    h0 = h0o;
    h1 = h1o;
  }

  // final gconv -> {'0': 32}
  edge_conv_kernel<16, 32, 1, 1, 1><<<EB, 32, 0, stream>>>(xer, K01, K11, h0, src, RAD_ARGS(4), fmsg, 32, 0, 0);
  edge_conv_kernel<16, 32, 1, 3, 1><<<EB, 32, 0, stream>>>(xer, K01, K11, h1, src, RAD_ARGS(10), fmsg, 32, 0, 1);
  hipMemsetAsync(agg, 0, (size_t)NN * 32 * sizeof(float), stream);
  edge_agg_kernel<<<(NE * 32 + T - 1) / T, T, 0, stream>>>(fmsg, dst, agg);
  self_kernel<<<(NN * 32 + T - 1) / T, T, 0, stream>>>(prm[16], h0, agg);
  hipMemsetAsync(gsum, 0, (size_t)(NG * 32 + NG) * sizeof(float), stream);
  pool_kernel<<<(NN * 32 + T - 1) / T, T, 0, stream>>>(agg, n2g, gsum, gcnt);
  head_kernel<<<1, NG, 0, stream>>>(gsum, gcnt, prm[0], prm[1], prm[2], prm[3], (float*)d_out);
}